// RewardPredictor_48498770706579
// MI455X (gfx1250) — compile-verified
//
#include <hip/hip_runtime.h>
#include <math.h>

// Problem sizes (fixed by the reference)
#define BB   256
#define NN   512
#define DIN  512
#define DKK  128
#define DVH  64      // DV / H
#define HH   8
#define DVV  512
#define DHH  1024

typedef __attribute__((ext_vector_type(2))) float v2f;
typedef __attribute__((ext_vector_type(8))) float v8f;

__device__ __forceinline__ v8f wmma4(v2f a, v2f b, v8f c) {
  // D = A(16x4) * B(4x16) + C(16x16), fp32
  return __builtin_amdgcn_wmma_f32_16x16x4_f32(false, a, false, b, (short)0, c,
                                               false, false);
}

// ---------------------------------------------------------------------------
// Kernel 1: qk[h,i] = sum_k w_k[h,i,k] * q[h,k]   (fold query into key proj)
// ---------------------------------------------------------------------------
__global__ void k_qkfold(const float* __restrict__ wk, const float* __restrict__ q,
                         float* __restrict__ qk) {
  int t = blockIdx.x * blockDim.x + threadIdx.x;   // 0 .. H*DIN-1
  int h = t >> 9;
  const float* wr = wk + (size_t)t * DKK;
  const float* qr = q + h * DKK;
  float acc = 0.f;
#pragma unroll 4
  for (int k = 0; k < DKK; ++k) acc += wr[k] * qr[k];
  qk[t] = acc;
}

// ---------------------------------------------------------------------------
// Kernel 2: logits[b,h,n] = (objects[b,n,:] . qk[h,:]) / sqrt(DK)
//   one wave per (b,n); qk staged in LDS; float4 streaming loads of objects
// ---------------------------------------------------------------------------
__global__ void k_logits(const float* __restrict__ objects,
                         const float* __restrict__ qk,
                         float* __restrict__ logits) {
  __shared__ float sqk[HH * DIN];                  // 16 KB
  for (int i = threadIdx.x; i < HH * DIN; i += blockDim.x) sqk[i] = qk[i];
  __syncthreads();

  int wave = threadIdx.x >> 5;
  int lane = threadIdx.x & 31;
  int bn   = blockIdx.x * 8 + wave;                // 0 .. B*N-1
  int b    = bn >> 9;
  int n    = bn & 511;

  const float4* orow = (const float4*)(objects + (size_t)bn * DIN);
  float acc[HH];
#pragma unroll
  for (int h = 0; h < HH; ++h) acc[h] = 0.f;

#pragma unroll
  for (int t = 0; t < 4; ++t) {
    int i4 = lane + 32 * t;                        // float4 index
    float4 o = orow[i4];
    int ib = i4 * 4;
#pragma unroll
    for (int h = 0; h < HH; ++h) {
      const float* qh = sqk + h * DIN + ib;
      acc[h] += o.x * qh[0] + o.y * qh[1] + o.z * qh[2] + o.w * qh[3];
    }
  }
  // wave32 reduction
#pragma unroll
  for (int off = 16; off > 0; off >>= 1) {
#pragma unroll
    for (int h = 0; h < HH; ++h) acc[h] += __shfl_xor(acc[h], off, 32);
  }
  if (lane == 0) {
    const float scale = 0.08838834764831845f;      // 1/sqrt(128)
#pragma unroll
    for (int h = 0; h < HH; ++h)
      logits[((size_t)b * HH + h) * NN + n] = acc[h] * scale;
  }
}

// ---------------------------------------------------------------------------
// Kernel 3: in-place masked softmax over n per (b,h)
//   max BEFORE mask, mask applied AFTER exp, denom + 1e-6 (matches reference)
// ---------------------------------------------------------------------------
__global__ void k_softmax(float* __restrict__ lw, const unsigned char* __restrict__ mask) {
  int bh = blockIdx.x;
  int b  = bh >> 3;
  float* row = lw + (size_t)bh * NN;
  const unsigned char* mrow = mask + (size_t)b * NN;
  __shared__ float red[256];
  int tid = threadIdx.x;

  float l0 = row[tid], l1 = row[tid + 256];
  red[tid] = fmaxf(l0, l1);
  __syncthreads();
  for (int s = 128; s > 0; s >>= 1) {
    if (tid < s) red[tid] = fmaxf(red[tid], red[tid + s]);
    __syncthreads();
  }
  float mx = red[0];
  __syncthreads();

  float e0 = __expf(l0 - mx) * (mrow[tid]       ? 1.f : 0.f);
  float e1 = __expf(l1 - mx) * (mrow[tid + 256] ? 1.f : 0.f);
  red[tid] = e0 + e1;
  __syncthreads();
  for (int s = 128; s > 0; s >>= 1) {
    if (tid < s) red[tid] += red[tid + s];
    __syncthreads();
  }
  float inv = 1.f / (red[0] + 1e-6f);
  row[tid]       = e0 * inv;
  row[tid + 256] = e1 * inv;
}

// ---------------------------------------------------------------------------
// Kernel 4 (WMMA, streams objects 2nd time):
//   s[b,h,i] = sum_n w[b,h,n] * objects[b,n,i]
//   A = weights padded [16 x 512] (rows 8..15 zero via 0/1 selector, loads
//   always in-bounds -> branch-free), B = objects[b] [512 x 512]
//   1 block per b, 8 waves x 4 N-tiles of 16 cover DIN=512
// ---------------------------------------------------------------------------
__global__ void k_wsum(const float* __restrict__ objects,
                       const float* __restrict__ weights,
                       float* __restrict__ s) {
  int b    = blockIdx.x;
  int wave = threadIdx.x >> 5;
  int lane = threadIdx.x & 31;
  int m    = lane & 15;
  int kh   = lane >> 4;                            // 0/1 -> K half
  int mrow = (m < HH) ? m : 0;                     // clamped (valid) row
  float sel = (m < HH) ? 1.f : 0.f;                // zero-pad rows 8..15

  const float2* wrow2 = (const float2*)(weights + ((size_t)b * HH + mrow) * NN);
  const float*  ob    = objects + (size_t)b * NN * DIN;

  v8f acc[4] = {};
  int ic[4];
#pragma unroll
  for (int t = 0; t < 4; ++t) ic[t] = (wave * 4 + t) * 16 + m;

  for (int k0 = 0; k0 < NN; k0 += 4) {
    int ka = k0 + kh * 2;                          // even
    float2 av = wrow2[ka >> 1];                    // one global_load_b64
    v2f a;
    a.x = av.x * sel;
    a.y = av.y * sel;
#pragma unroll
    for (int t = 0; t < 4; ++t) {
      v2f bf;
      bf.x = ob[(size_t)ka * DIN + ic[t]];
      bf.y = ob[(size_t)(ka + 1) * DIN + ic[t]];
      acc[t] = wmma4(a, bf, acc[t]);
    }
  }
  if (lane < 16) {                                 // lanes 0..15 hold M=0..7 (valid heads)
#pragma unroll
    for (int t = 0; t < 4; ++t) {
      int i = (wave * 4 + t) * 16 + lane;
#pragma unroll
      for (int r = 0; r < HH; ++r)
        s[((size_t)b * HH + r) * DIN + i] = acc[t][r];
    }
  }
}

// ---------------------------------------------------------------------------
// Kernel 5 (WMMA): pooled[b, h*64+d] = sum_i s[b,h,i] * w_v[h,i,d]
//   per head: [B x 512] x [512 x 64]; grid (16 b-tiles, 8 heads), 4 waves = 4 d-tiles
// ---------------------------------------------------------------------------
__global__ void k_pool(const float* __restrict__ s, const float* __restrict__ wv,
                       float* __restrict__ pooled) {
  int mt   = blockIdx.x;                           // b tile
  int h    = blockIdx.y;
  int wave = threadIdx.x >> 5;                     // d tile
  int lane = threadIdx.x & 31;
  int m    = lane & 15;
  int kh   = lane >> 4;
  int bbase = mt * 16, nbase = wave * 16;

  const float2* srow2 = (const float2*)(s + ((size_t)(bbase + m) * HH + h) * DIN);

  v8f acc = {};
  for (int k0 = 0; k0 < DIN; k0 += 4) {
    int ka = k0 + kh * 2;                          // even
    float2 av = srow2[ka >> 1];                    // one global_load_b64
    v2f a, bf;
    a.x  = av.x;
    a.y  = av.y;
    bf.x = wv[((size_t)h * DIN + ka)     * DVH + nbase + m];
    bf.y = wv[((size_t)h * DIN + ka + 1) * DVH + nbase + m];
    acc = wmma4(a, bf, acc);
  }
#pragma unroll
  for (int r = 0; r < 8; ++r) {
    int row = r + 8 * kh;
    pooled[(size_t)(bbase + row) * DVV + h * DVH + nbase + m] = acc[r];
  }
}

// ---------------------------------------------------------------------------
// Kernel 6 (WMMA): h1 = leaky_relu(pooled @ W1 + b1)   [256,512]x[512,1024]
// ---------------------------------------------------------------------------
__global__ void k_mlp1(const float* __restrict__ pooled, const float* __restrict__ W1,
                       const float* __restrict__ b1, float* __restrict__ h1) {
  int mt   = blockIdx.x;                           // 0..15
  int wave = threadIdx.x >> 5;
  int lane = threadIdx.x & 31;
  int nt   = blockIdx.y * 8 + wave;                // 0..63
  int m    = lane & 15;
  int kh   = lane >> 4;
  int mbase = mt * 16, nbase = nt * 16;

  const float2* prow2 = (const float2*)(pooled + (size_t)(mbase + m) * DVV);

  v8f acc = {};
  for (int k0 = 0; k0 < DVV; k0 += 4) {
    int ka = k0 + kh * 2;                          // even
    float2 av = prow2[ka >> 1];                    // one global_load_b64
    v2f a, bf;
    a.x  = av.x;
    a.y  = av.y;
    bf.x = W1[(size_t)ka       * DHH + nbase + m];
    bf.y = W1[(size_t)(ka + 1) * DHH + nbase + m];
    acc = wmma4(a, bf, acc);
  }
  float bias = b1[nbase + m];
#pragma unroll
  for (int r = 0; r < 8; ++r) {
    int row = r + 8 * kh;
    float v = acc[r] + bias;
    v = (v > 0.f) ? v : 0.01f * v;                 // leaky_relu, slope 0.01
    h1[(size_t)(mbase + row) * DHH + nbase + m] = v;
  }
}

// ---------------------------------------------------------------------------
// Kernel 7: out[b] = h1[b,:] . W2 + b2
// ---------------------------------------------------------------------------
__global__ void k_out(const float* __restrict__ h1, const float* __restrict__ W2,
                      const float* __restrict__ b2, float* __restrict__ out) {
  int b = blockIdx.x, tid = threadIdx.x;
  const float* row = h1 + (size_t)b * DHH;
  float acc = 0.f;
  for (int j = tid; j < DHH; j += 256) acc += row[j] * W2[j];
  __shared__ float red[256];
  red[tid] = acc;
  __syncthreads();
  for (int s = 128; s > 0; s >>= 1) {
    if (tid < s) red[tid] += red[tid + s];
    __syncthreads();
  }
  if (tid == 0) out[b] = red[0] + b2[0];
}

// ---------------------------------------------------------------------------
extern "C" void kernel_launch(void* const* d_in, const int* in_sizes, int n_in,
                              void* d_out, int out_size, void* d_ws, size_t ws_size,
                              hipStream_t stream) {
  const float*         objects = (const float*)d_in[0];
  const unsigned char* mask    = (const unsigned char*)d_in[1];  // jax bool = 1 byte
  const float*         q       = (const float*)d_in[2];
  const float*         wk      = (const float*)d_in[3];
  const float*         wv      = (const float*)d_in[4];
  const float*         W1      = (const float*)d_in[5];
  const float*         b1      = (const float*)d_in[6];
  const float*         W2      = (const float*)d_in[7];
  const float*         b2      = (const float*)d_in[8];
  float* out = (float*)d_out;

  float* ws     = (float*)d_ws;
  float* qk     = ws;                        // H*DIN            =    4096
  float* lw     = qk + HH * DIN;             // B*H*N (logits -> weights, in place)
  float* s      = lw + (size_t)BB * HH * NN; // B*H*DIN          = 1048576
  float* pooled = s  + (size_t)BB * HH * DIN;// B*DV             =  131072
  float* h1     = pooled + (size_t)BB * DVV; // B*DH             =  262144

  k_qkfold <<<(HH * DIN) / 256, 256, 0, stream>>>(wk, q, qk);
  k_logits <<<(BB * NN) / 8,    256, 0, stream>>>(objects, qk, lw);
  k_softmax<<<BB * HH,          256, 0, stream>>>(lw, mask);
  k_wsum   <<<BB,               256, 0, stream>>>(objects, lw, s);
  k_pool   <<<dim3(BB / 16, HH),128, 0, stream>>>(s, wv, pooled);
  k_mlp1   <<<dim3(BB / 16, 8), 256, 0, stream>>>(pooled, W1, b1, h1);
  k_out    <<<BB,               256, 0, stream>>>(h1, W2, b2, out);
}